// LagNet_65498251264054
// MI455X (gfx1250) — compile-verified
//
#include <hip/hip_runtime.h>
#include <math.h>

// ---------------------------------------------------------------------------
// LagNet on MI455X (gfx1250, wave32).
// Pipeline (stream-ordered, graph-capturable):
//   1. h1/s1, g1 matvecs (W [512,32])            -- wave-per-row
//   2. dh1 = s1[:,None]*W1                        -- elementwise
//   3. h2/s2, g2 matvecs (W [512,512])
//   4. h3/s3, gv matvecs
//   5. l_diag (exp head), l_off heads
//   6. M  = diag(s2) * (W2 @ dh1)   via V_WMMA_F32_16X16X4_F32
//   7. N  = diag(s3) * (W3 @ M)     via WMMA
//   8. dl = [diag(l_diag)*(Wd@N) ; Wo@N]  via WMMA (two launches)
//   9. finalize: v = dl@q_t, assemble L/Dqt/Fjk, 32x32 contractions -> tau
// All fp32 (exact vs reference up to reduction order).
// ---------------------------------------------------------------------------

typedef __attribute__((ext_vector_type(2))) float v2f;
typedef __attribute__((ext_vector_type(8))) float v8f;

#define QDIM 32
#define HID  512
#define NOFF 496

// workspace layout (float offsets)
#define WS_H1   0
#define WS_S1   512
#define WS_H2   1024
#define WS_S2   1536
#define WS_H3   2048
#define WS_S3   2560
#define WS_G1   3072
#define WS_G2   3584
#define WS_GV   4096
#define WS_LD   4128
#define WS_LO   4160
#define WS_DH1  4656
#define WS_M    21040
#define WS_N    37424
#define WS_DL   53808
// end: 70704 floats (~283 KB)

// mode: 0 = sigmoid (+optional s=y(1-y)), 1 = exp, 2 = linear
__global__ void matvec_act(const float* __restrict__ W, const float* __restrict__ x,
                           const float* __restrict__ bias, float* __restrict__ y,
                           float* __restrict__ s, int rows, int cols, int mode) {
    const int wave = threadIdx.x >> 5;
    const int r = blockIdx.x * (blockDim.x >> 5) + wave;   // wave-uniform
    if (r >= rows) return;
    const int lane = threadIdx.x & 31;
    float sum = 0.0f;
    for (int c = lane; c < cols; c += 32)
        sum += W[r * cols + c] * x[c];
    #pragma unroll
    for (int m = 16; m >= 1; m >>= 1)
        sum += __shfl_xor(sum, m, 32);
    if (lane == 0) {
        const float t = sum + bias[r];
        if (mode == 0) {
            const float v = 1.0f / (1.0f + expf(-t));
            y[r] = v;
            if (s) s[r] = v * (1.0f - v);
        } else if (mode == 1) {
            y[r] = expf(t);
        } else {
            y[r] = t;
        }
    }
}

// dh1[c,k] = s1[c] * W1[c,k]   (512*32 elements)
__global__ void rowscale32(const float* __restrict__ W1, const float* __restrict__ s1,
                           float* __restrict__ dh1) {
    const int t = blockIdx.x * blockDim.x + threadIdx.x;
    dh1[t] = s1[t >> 5] * W1[t];
}

// C[r,n] = (scale ? scale[r] : 1) * sum_k A[r,k]*B[k,n]
// A row-major [16*rowTiles, 512], B [512, 32], C [16*rowTiles, 32].
// One wave per 16x16 output tile; K-loop of 128 x V_WMMA_F32_16X16X4_F32.
__global__ void wmma_gemm(const float* __restrict__ A, const float* __restrict__ B,
                          const float* __restrict__ scale, float* __restrict__ C,
                          int rowTiles) {
    const int wave  = threadIdx.x >> 5;
    const int tile  = blockIdx.x * 8 + wave;               // wave-uniform
    const int nTile = rowTiles * 2;
    if (tile >= nTile) return;                             // whole-wave exit: EXEC all 1s below
    const int tm = tile >> 1;
    const int tn = tile & 1;
    const int lane = threadIdx.x & 31;
    const int half = lane >> 4;                            // K sub-pair select
    const int l15  = lane & 15;

    // A lane layout (16x4 f32): lane<16 -> row l15, K {k0,k0+1}; lane>=16 -> K {k0+2,k0+3}
    const float* Arow = A + (tm * 16 + l15) * HID + 2 * half;
    // B lane layout (4x16 f32): lane<16 -> col l15, K {k0,k0+1}; lane>=16 -> K {k0+2,k0+3}
    const int col = tn * 16 + l15;
    const float* Bcol = B + 2 * half * QDIM + col;

    v8f acc = {0.f, 0.f, 0.f, 0.f, 0.f, 0.f, 0.f, 0.f};
    #pragma unroll 8
    for (int k0 = 0; k0 < HID; k0 += 4) {
        const v2f a = *reinterpret_cast<const v2f*>(Arow + k0);   // 8B-aligned pair
        v2f b;
        b.x = Bcol[k0 * QDIM];
        b.y = Bcol[k0 * QDIM + QDIM];
        acc = __builtin_amdgcn_wmma_f32_16x16x4_f32(
            /*neg_a=*/false, a, /*neg_b=*/false, b,
            /*c_mod=*/(short)0, acc, /*reuse_a=*/false, /*reuse_b=*/false);
    }

    // C/D layout: vgpr i -> row tm*16 + i + (lane>=16 ? 8 : 0), col = tn*16 + l15
    const int rbase = tm * 16 + (half ? 8 : 0);
    // Hoist the (wave-uniform) scale decision so the epilogue is branch-free
    float sc[8];
    if (scale != nullptr) {
        #pragma unroll
        for (int i = 0; i < 8; ++i) sc[i] = scale[rbase + i];
    } else {
        #pragma unroll
        for (int i = 0; i < 8; ++i) sc[i] = 1.0f;
    }
    #pragma unroll
    for (int i = 0; i < 8; ++i)
        C[(rbase + i) * QDIM + col] = acc[i] * sc[i];
}

// Final assembly: v = dl@q_t; build L, Dqt, Fjk; contract to tau.
// Note comp_5 == comp_3 == Dqt @ (L^T q_t), so tail = L@(u1+w2) + 0.5*c3 - 0.5*c4 + g.
__global__ void finalize(const float* __restrict__ q_t, const float* __restrict__ q_tt,
                         const float* __restrict__ l_diag, const float* __restrict__ l_off,
                         const float* __restrict__ dl, const float* __restrict__ gv,
                         float* __restrict__ out) {
    __shared__ float Ls[QDIM][QDIM];
    __shared__ float Dq[QDIM][QDIM];
    __shared__ float Fk[QDIM][QDIM];
    __shared__ float vv[QDIM + NOFF];
    __shared__ float LTqt[QDIM], u1[QDIM], w2[QDIM];
    __shared__ float qts[QDIM], qtts[QDIM];

    const int t = threadIdx.x;  // 1024 threads
    if (t < QDIM) { qts[t] = q_t[t]; qtts[t] = q_tt[t]; }
    if (t < QDIM + NOFF) {
        float sum = 0.0f;
        #pragma unroll
        for (int k = 0; k < QDIM; ++k) sum += dl[t * QDIM + k] * q_t[k];
        vv[t] = sum;
    }
    __syncthreads();

    {
        const int a = t >> 5, b = t & 31;
        float lij, dij;
        if (a == b)      { lij = l_diag[a]; dij = vv[a]; }
        else if (a > b)  { const int m = (a * (a - 1)) / 2 + b; lij = l_off[m]; dij = vv[QDIM + m]; }
        else             { lij = 0.0f; dij = 0.0f; }
        Ls[a][b] = lij;
        Dq[a][b] = dij;
        // Fjk[a,b] = sum_i dL_dq[i,a,b]*q_t[i]: i==a diag term + strictly-lower (i>a, col a)
        float f = qts[a] * dl[a * QDIM + b];
        for (int r = a + 1; r < QDIM; ++r)
            f += qts[r] * dl[(QDIM + (r * (r - 1)) / 2 + a) * QDIM + b];
        Fk[a][b] = f;
    }
    __syncthreads();

    if (t < QDIM) {
        float a0 = 0.f, a1 = 0.f, a2 = 0.f;
        #pragma unroll
        for (int j = 0; j < QDIM; ++j) {
            a0 += Ls[j][t] * qts[j];    // LTqt = L^T q_t
            a1 += Ls[j][t] * qtts[j];   // u1   = L^T q_tt
            a2 += Dq[j][t] * qts[j];    // w2   = Dqt^T q_t
        }
        LTqt[t] = a0; u1[t] = a1; w2[t] = a2;
    }
    __syncthreads();

    if (t < QDIM) {
        float c12 = 0.f, c3 = 0.f, c4 = 0.f;
        #pragma unroll
        for (int k = 0; k < QDIM; ++k) {
            c12 += Ls[t][k] * (u1[k] + w2[k]);   // comp_1 + comp_2
            c3  += Dq[t][k] * LTqt[k];           // comp_3 (== comp_5)
            c4  += Fk[t][k] * LTqt[k];           // comp_4
        }
        out[t] = c12 + 0.5f * c3 - 0.5f * c4 + gv[t];
    }
}

extern "C" void kernel_launch(void* const* d_in, const int* in_sizes, int n_in,
                              void* d_out, int out_size, void* d_ws, size_t ws_size,
                              hipStream_t stream) {
    (void)in_sizes; (void)n_in; (void)out_size; (void)ws_size;
    const float* q    = (const float*)d_in[0];
    const float* q_t  = (const float*)d_in[1];
    const float* q_tt = (const float*)d_in[2];
    const float* W1   = (const float*)d_in[3];
    const float* b1   = (const float*)d_in[4];
    const float* W2   = (const float*)d_in[5];
    const float* b2   = (const float*)d_in[6];
    const float* W3   = (const float*)d_in[7];
    const float* b3   = (const float*)d_in[8];
    const float* Wd   = (const float*)d_in[9];
    const float* bd   = (const float*)d_in[10];
    const float* Wo   = (const float*)d_in[11];
    const float* bo   = (const float*)d_in[12];
    const float* Wg1  = (const float*)d_in[13];
    const float* bg1  = (const float*)d_in[14];
    const float* Wg2  = (const float*)d_in[15];
    const float* bg2  = (const float*)d_in[16];
    const float* Wg3  = (const float*)d_in[17];
    const float* bg3  = (const float*)d_in[18];

    float* ws  = (float*)d_ws;
    float* out = (float*)d_out;

    // Stage 1: h1/s1, g1 (cols=32), dh1
    matvec_act<<<64, 256, 0, stream>>>(W1,  q, b1,  ws + WS_H1, ws + WS_S1, HID, QDIM, 0);
    matvec_act<<<64, 256, 0, stream>>>(Wg1, q, bg1, ws + WS_G1, nullptr,    HID, QDIM, 0);
    rowscale32<<<64, 256, 0, stream>>>(W1, ws + WS_S1, ws + WS_DH1);

    // Stage 2: h2/s2, g2
    matvec_act<<<64, 256, 0, stream>>>(W2,  ws + WS_H1, b2,  ws + WS_H2, ws + WS_S2, HID, HID, 0);
    matvec_act<<<64, 256, 0, stream>>>(Wg2, ws + WS_G1, bg2, ws + WS_G2, nullptr,    HID, HID, 0);

    // Stage 3: h3/s3, gv
    matvec_act<<<64, 256, 0, stream>>>(W3,  ws + WS_H2, b3,  ws + WS_H3, ws + WS_S3, HID, HID, 0);
    matvec_act<<<4,  256, 0, stream>>>(Wg3, ws + WS_G2, bg3, ws + WS_GV, nullptr,    QDIM, HID, 2);

    // Heads: l_diag (exp), l_off (linear)
    matvec_act<<<4,  256, 0, stream>>>(Wd, ws + WS_H3, bd, ws + WS_LD, nullptr, QDIM, HID, 1);
    matvec_act<<<62, 256, 0, stream>>>(Wo, ws + WS_H3, bo, ws + WS_LO, nullptr, NOFF, HID, 2);

    // Jacobian GEMM chain (fp32 WMMA)
    wmma_gemm<<<8, 256, 0, stream>>>(W2, ws + WS_DH1, ws + WS_S2, ws + WS_M, 32);
    wmma_gemm<<<8, 256, 0, stream>>>(W3, ws + WS_M,   ws + WS_S3, ws + WS_N, 32);
    wmma_gemm<<<1, 256, 0, stream>>>(Wd, ws + WS_N,   ws + WS_LD, ws + WS_DL,            2);
    wmma_gemm<<<8, 256, 0, stream>>>(Wo, ws + WS_N,   nullptr,    ws + WS_DL + QDIM*QDIM, 31);

    // Final 32-wide assembly + contraction
    finalize<<<1, 1024, 0, stream>>>(q_t, q_tt, ws + WS_LD, ws + WS_LO,
                                     ws + WS_DL, ws + WS_GV, out);
}